// mTAN_28381143891995
// MI455X (gfx1250) — compile-verified
//
#include <hip/hip_runtime.h>
#include <hip/hip_bf16.h>

// ---- problem constants (match reference) ----
#define NB   4
#define NT   512
#define NH   8
#define HD   16
#define HID  128
#define XLD  129               // inputs row stride (D+1)
#define SCL  0.25f             // HD^-0.5
#define SPAD 516               // LDS score-row stride (bank-conflict pad)
#define KPAD 20                // K/V LDS row stride: 16B-aligned AND conflict-free

typedef __attribute__((ext_vector_type(2))) float v2f;
typedef __attribute__((ext_vector_type(8))) float v8f;

// V_WMMA_F32_16X16X4_F32:  D(16x16 f32) = A(16x4 f32) * B(4x16 f32) + C
__device__ __forceinline__ v8f wmma4(v2f a, v2f b, v8f c) {
  return __builtin_amdgcn_wmma_f32_16x16x4_f32(false, a, false, b, (short)0, c,
                                               false, false);
}

#if __has_builtin(__builtin_amdgcn_tanhf)
__device__ __forceinline__ float fast_tanh(float x) {
  return __builtin_amdgcn_tanhf(x);        // v_tanh_f32 (CDNA5 TRANS op)
}
#else
__device__ __forceinline__ float fast_tanh(float x) {
  float y;
  asm volatile("v_tanh_f32 %0, %1" : "=v"(y) : "v"(x));
  return y;
}
#endif

// Issue async global->LDS copy of a 512x16 f32 head slice (row stride HID)
// into dst[512][KPAD], 16B chunks, coalesced.  Tracked by ASYNCcnt.
__device__ __forceinline__ void stage_head_issue(const float* __restrict__ src,
                                                 float (*dst)[KPAD], int tid) {
  for (int i = tid; i < 2048; i += 128) {
    const int r  = i >> 2;
    const int c4 = (i & 3) * 4;
    const unsigned lds = (unsigned)(uintptr_t)&dst[r][c4];       // LDS offset
    const unsigned long long ga =
        (unsigned long long)(uintptr_t)(src + (size_t)r * HID + c4);
    asm volatile("global_load_async_to_lds_b128 %0, %1, off"
                 :: "v"(lds), "v"(ga) : "memory");
  }
}
__device__ __forceinline__ void stage_wait() {
  asm volatile("s_wait_asynccnt 0x0" ::: "memory");
}

// -------------------------------------------------------------------------
// Kernel 1: Q/K/V projections.  grid = (rowTiles=128, colTiles=8, 3 mats),
// one wave per 16x16 output tile, K accumulated in 32 WMMA steps of K=4.
// -------------------------------------------------------------------------
__global__ void qkv_proj(const float* __restrict__ inp,
                         const float* __restrict__ Wq,
                         const float* __restrict__ Wk,
                         const float* __restrict__ Wv,
                         float* __restrict__ ws) {
  const int rt   = blockIdx.x;                  // row tile over B*T
  const int ct   = blockIdx.y;                  // col tile over HID
  const int wsel = blockIdx.z;                  // 0:Q 1:K 2:V
  const float* W = (wsel == 0) ? Wq : (wsel == 1) ? Wk : Wv;
  float* out = ws + (size_t)wsel * (NB * NT * HID);

  const int lane = threadIdx.x & 31;
  const int m    = lane & 15;                   // A row / B,D col
  const int hi   = lane >> 4;                   // 0: K=0,1  1: K=2,3

  const float* xrow = inp + (size_t)(rt * 16 + m) * XLD;
  const float* wcol = W + ct * 16 + m;
  __builtin_prefetch(wcol, 0, 1);               // global_prefetch_b8

  v8f acc = {};
#pragma unroll
  for (int kk = 0; kk < HID; kk += 4) {
    v2f a, bm;
    a.x  = xrow[kk + 2 * hi];
    a.y  = xrow[kk + 2 * hi + 1];
    bm.x = wcol[(size_t)(kk + 2 * hi) * HID];
    bm.y = wcol[(size_t)(kk + 2 * hi + 1) * HID];
    acc  = wmma4(a, bm, acc);
  }
#pragma unroll
  for (int r = 0; r < 8; ++r)
    out[(size_t)(rt * 16 + r + hi * 8) * HID + ct * 16 + m] = acc[r];
}

// -------------------------------------------------------------------------
// Kernel 2: fused temporal attention for one (b, h, q-tile of 16).
// 4 waves / block.  K slice async-staged to LDS; V overlapped with softmax.
//   scores = WMMA(Q,K^T) + fused tanh time-term  ->  LDS (16 x 512)
//   softmax in LDS  ->  attn tile = WMMA(w, V) with cross-wave k-split
// time_enc is NEVER materialized (saves 17 GB of HBM traffic).
// -------------------------------------------------------------------------
__global__ void attention_core(const float* __restrict__ inp,
                               const unsigned char* __restrict__ mask,
                               const float* __restrict__ Wt,
                               const float* __restrict__ bt,
                               const float* __restrict__ ws,
                               float* __restrict__ attn) {
  __shared__ float s_sc[16][SPAD];              // scores, later attn partials
  __shared__ float s_k[NT][KPAD];               // K slice, then V slice
  __shared__ float s_q[16][17];
  __shared__ float s_tq[16];
  __shared__ float s_wt[16];
  __shared__ float s_bt[16];

  const int qt = blockIdx.x, h = blockIdx.y, b = blockIdx.z;
  const int q0 = qt * 16;
  const float* Qf = ws;
  const float* Kf = ws + (size_t)(NB * NT * HID);
  const float* Vf = ws + (size_t)(2 * NB * NT * HID);
  const float* Khead = Kf + (size_t)(b * NT) * HID + h * HD;
  const float* Vhead = Vf + (size_t)(b * NT) * HID + h * HD;

  const int tid  = threadIdx.x;                 // 0..127
  const int wv   = tid >> 5;                    // wave 0..3
  const int lane = tid & 31;
  const int m    = lane & 15;
  const int hi   = lane >> 4;

  // async-stage the K head slice while we fill Q/params
  stage_head_issue(Khead, s_k, tid);
  for (int i = tid; i < 256; i += 128) {
    const int q = i >> 4, d = i & 15;
    s_q[q][d] = Qf[(size_t)(b * NT + q0 + q) * HID + h * HD + d];
  }
  if (tid < 16) {
    s_tq[tid] = inp[(size_t)(b * NT + q0 + tid) * XLD + HID];
    s_wt[tid] = Wt[h * HD + tid];
    s_bt[tid] = bt[h * HD + tid];
  }
  stage_wait();
  __syncthreads();

  // ---- pass 1: scores; wave wv owns k-tiles wv, wv+4, ... ----
  for (int kt = wv; kt < NT / 16; kt += 4) {
    const int k0 = kt * 16;
    v8f s = {};
#pragma unroll
    for (int dd = 0; dd < HD; dd += 4) {
      v2f a, bm;
      a.x  = s_q[m][dd + 2 * hi];
      a.y  = s_q[m][dd + 2 * hi + 1];
      bm.x = s_k[k0 + m][dd + 2 * hi];
      bm.y = s_k[k0 + m][dd + 2 * hi + 1];
      s    = wmma4(a, bm, s);
    }
    const float tk = inp[(size_t)(b * NT + k0 + m) * XLD + HID];
    const float mk = mask[b * NT + k0 + m] ? 0.0f : -1e9f;
#pragma unroll
    for (int r = 0; r < 8; ++r) {
      const int   q     = r + hi * 8;            // C/D row mapping
      const float delta = s_tq[q] - tk;
      float dot = 0.0f;
#pragma unroll
      for (int d = 0; d < HD; ++d)
        dot += s_q[q][d] * fast_tanh(delta * s_wt[d] + s_bt[d]);
      s_sc[q][k0 + m] = (s[r] + dot) * SCL + mk;
    }
  }
  __syncthreads();

  // overlap: refill s_k with the V head slice while softmax runs
  stage_head_issue(Vhead, s_k, tid);

  // ---- softmax over k; wave wv owns rows 4*wv .. 4*wv+3 ----
  for (int row = wv * 4; row < wv * 4 + 4; ++row) {
    float mx = -3.4e38f;
    for (int c = lane; c < NT; c += 32) mx = fmaxf(mx, s_sc[row][c]);
#pragma unroll
    for (int off = 16; off > 0; off >>= 1)
      mx = fmaxf(mx, __shfl_xor(mx, off, 32));
    float sum = 0.0f;
    for (int c = lane; c < NT; c += 32) {
      const float e = __expf(s_sc[row][c] - mx);
      s_sc[row][c]  = e;
      sum += e;
    }
#pragma unroll
    for (int off = 16; off > 0; off >>= 1) sum += __shfl_xor(sum, off, 32);
    const float inv = 1.0f / sum;
    for (int c = lane; c < NT; c += 32) s_sc[row][c] *= inv;
  }
  stage_wait();
  __syncthreads();

  // ---- pass 2: attn = w @ V, wave wv handles k in [wv*128, wv*128+128) ----
  v8f o = {};
  const int kb = wv * 128;
#pragma unroll
  for (int kk = 0; kk < 128; kk += 4) {
    v2f a, bm;
    a.x  = s_sc[m][kb + kk + 2 * hi];
    a.y  = s_sc[m][kb + kk + 2 * hi + 1];
    bm.x = s_k[kb + kk + 2 * hi][m];
    bm.y = s_k[kb + kk + 2 * hi + 1][m];
    o    = wmma4(a, bm, o);
  }
  __syncthreads();                              // all s_sc reads done
  float* part = &s_sc[0][0];                    // reuse as 4x(16x16) partials
#pragma unroll
  for (int r = 0; r < 8; ++r)
    part[wv * 256 + (r + hi * 8) * 16 + m] = o[r];
  __syncthreads();
  for (int i = tid; i < 256; i += 128) {
    const int q = i >> 4, n = i & 15;
    const float v = part[q * 16 + n] + part[256 + q * 16 + n] +
                    part[512 + q * 16 + n] + part[768 + q * 16 + n];
    attn[(size_t)(b * NT + q0 + q) * HID + h * HD + n] = v;
  }
}

// -------------------------------------------------------------------------
// Kernel 3: out = attn @ Wo + bo; y = x + out; LayerNorm.
// 8 waves per block: wave w owns col-tile w (WMMA); LN stats via LDS.
// -------------------------------------------------------------------------
__global__ void outproj_ln(const float* __restrict__ inp,
                           const float* __restrict__ attn,
                           const float* __restrict__ Wo,
                           const float* __restrict__ bo,
                           const float* __restrict__ gamma,
                           const float* __restrict__ beta,
                           float* __restrict__ y) {
  __shared__ float s_y[16][HID + 1];
  const int rt   = blockIdx.x;
  const int wv   = threadIdx.x >> 5;            // wave id == col tile
  const int lane = threadIdx.x & 31;
  const int m    = lane & 15;
  const int hi   = lane >> 4;

  v8f acc = {};
#pragma unroll
  for (int kk = 0; kk < HID; kk += 4) {
    v2f a, bm;
    a.x  = attn[(size_t)(rt * 16 + m) * HID + kk + 2 * hi];
    a.y  = attn[(size_t)(rt * 16 + m) * HID + kk + 2 * hi + 1];
    bm.x = Wo[(size_t)(kk + 2 * hi) * HID + wv * 16 + m];
    bm.y = Wo[(size_t)(kk + 2 * hi + 1) * HID + wv * 16 + m];
    acc  = wmma4(a, bm, acc);
  }
#pragma unroll
  for (int r = 0; r < 8; ++r) {
    const int q   = r + hi * 8;
    const int col = wv * 16 + m;
    s_y[q][col] = acc[r] + bo[col] + inp[(size_t)(rt * 16 + q) * XLD + col];
  }
  __syncthreads();

  const int row    = threadIdx.x >> 4;          // 0..15
  const int within = threadIdx.x & 15;
  float s1 = 0.f, s2 = 0.f;
  for (int c = within; c < HID; c += 16) {
    const float v = s_y[row][c];
    s1 += v;
    s2 += v * v;
  }
#pragma unroll
  for (int off = 8; off > 0; off >>= 1) {       // 16-lane groups, half-wave aligned
    s1 += __shfl_xor(s1, off, 32);
    s2 += __shfl_xor(s2, off, 32);
  }
  const float mu  = s1 * (1.0f / HID);
  const float var = s2 * (1.0f / HID) - mu * mu;
  const float inv = rsqrtf(var + 1e-6f);
  for (int c = within; c < HID; c += 16)
    y[(size_t)(rt * 16 + row) * HID + c] =
        (s_y[row][c] - mu) * inv * gamma[c] + beta[c];
}

// -------------------------------------------------------------------------
// Kernel 4: masked mean over T -> context (B, HID)
// -------------------------------------------------------------------------
__global__ void context_reduce(const float* __restrict__ y,
                               const unsigned char* __restrict__ mask,
                               float* __restrict__ out) {
  const int b = blockIdx.x;
  const int j = threadIdx.x;
  float s = 0.f, cnt = 0.f;
  for (int t = 0; t < NT; ++t) {
    const float mm = mask[b * NT + t] ? 1.0f : 0.0f;
    s   += y[(size_t)(b * NT + t) * HID + j] * mm;
    cnt += mm;
  }
  out[b * HID + j] = s / (cnt + 1e-8f);
}

// -------------------------------------------------------------------------
extern "C" void kernel_launch(void* const* d_in, const int* in_sizes, int n_in,
                              void* d_out, int out_size, void* d_ws,
                              size_t ws_size, hipStream_t stream) {
  const float*         inputs = (const float*)d_in[0];
  const unsigned char* mask   = (const unsigned char*)d_in[1];
  const float*         Wq     = (const float*)d_in[2];
  const float*         Wk     = (const float*)d_in[3];
  const float*         Wv     = (const float*)d_in[4];
  const float*         Wt     = (const float*)d_in[5];
  const float*         bt     = (const float*)d_in[6];
  const float*         Wo     = (const float*)d_in[7];
  const float*         bo     = (const float*)d_in[8];
  const float*         gamma  = (const float*)d_in[9];
  const float*         beta   = (const float*)d_in[10];

  float* ws   = (float*)d_ws;                   // 5 x 1MB f32 buffers
  const size_t SEG = (size_t)NB * NT * HID;     // 262144
  float* attn = ws + 3 * SEG;
  float* ybuf = ws + 4 * SEG;

  qkv_proj<<<dim3(NB * NT / 16, HID / 16, 3), 32, 0, stream>>>(
      inputs, Wq, Wk, Wv, ws);
  attention_core<<<dim3(NT / 16, NH, NB), 128, 0, stream>>>(
      inputs, mask, Wt, bt, ws, attn);
  outproj_ln<<<dim3(NB * NT / 16), 256, 0, stream>>>(
      inputs, attn, Wo, bo, gamma, beta, ybuf);
  context_reduce<<<dim3(NB), HID, 0, stream>>>(ybuf, mask, (float*)d_out);
}